// RidgeRegression_69157563400904
// MI455X (gfx1250) — compile-verified
//
#include <hip/hip_runtime.h>
#include <hip/hip_bf16.h>

// ---------------------------------------------------------------------------
// Problem constants (from the reference)
// ---------------------------------------------------------------------------
#define S_DIM   4
#define B_DIM   64
#define K_DIM   40000
#define OUT_DIM 768
#define V_DIM   (64 * 64 * 48)

// workspace layout (ints at base, gathered matrix after 1 KiB)
#define META_COUNT 0    // [4]  samples per subject
#define META_START 4    // [4]  exclusive prefix of counts (row offset)
#define META_PERM  8    // [64] subject-sorted row -> original sample index
#define META_SUBJ  72   // [64] subject-sorted row -> subject id
#define GP_OFFSET_BYTES 1024

typedef __attribute__((ext_vector_type(2))) float v2f;
typedef __attribute__((ext_vector_type(8))) float v8f;

// ---------------------------------------------------------------------------
// Kernel 1: group the 64 samples by subject (order-preserving), serially.
// ---------------------------------------------------------------------------
__global__ void group_kernel(const int* __restrict__ id, int* __restrict__ meta) {
    if (threadIdx.x != 0) return;
    int cnt[S_DIM] = {0, 0, 0, 0};
    for (int i = 0; i < B_DIM; ++i) cnt[id[i] & 3]++;
    int st[S_DIM];
    st[0] = 0;
    for (int s = 1; s < S_DIM; ++s) st[s] = st[s - 1] + cnt[s - 1];
    for (int s = 0; s < S_DIM; ++s) {
        meta[META_COUNT + s] = cnt[s];
        meta[META_START + s] = st[s];
    }
    int ofs[S_DIM] = {st[0], st[1], st[2], st[3]};
    for (int i = 0; i < B_DIM; ++i) {
        int s = id[i] & 3;
        int r = ofs[s]++;
        meta[META_PERM + r] = i;
        meta[META_SUBJ + r] = s;
    }
}

// ---------------------------------------------------------------------------
// Kernel 2: dense gather. Gp[r, k] = fmri[sample(r), idx[subj(r), k]]
// Gp is subject-sorted, row-major [64, K] f32 (10.24 MB, stays in L2).
// ---------------------------------------------------------------------------
__global__ __launch_bounds__(256) void gather_kernel(const float* __restrict__ fmri,
                                                     const int*   __restrict__ idx,
                                                     const int*   __restrict__ meta,
                                                     float*       __restrict__ Gp) {
    const int k = blockIdx.x * 256 + threadIdx.x;
    if (k >= K_DIM) return;
    const int r = blockIdx.y;
    const int s = meta[META_SUBJ + r];
    const int b = meta[META_PERM + r];
    const int v = idx[s * K_DIM + k];
    Gp[(size_t)r * K_DIM + k] = fmri[(size_t)b * V_DIM + v];
}

// ---------------------------------------------------------------------------
// Kernel 3: per-subject GEMM with V_WMMA_F32_16X16X4_F32, specialized on the
// number of M-tiles (MT). All four MT variants are launched; each block
// checks the device-computed tile count and exits unless it matches.
//
// Grid: x = 48 output tiles of 16, y = subject. Block = 8 waves (wave32);
// wave w owns K positions k = 4*w + 32*i. Literal trip count (SALU loop
// control), 32-bit offset + imm addressing (GVS loads), and an explicit
// two-buffer software pipeline: chunk i+1's loads are issued before chunk i
// is drained, so the oldest-load wait always has a full chunk in flight.
// ---------------------------------------------------------------------------
template <int MT>
__global__ __launch_bounds__(256) void gemm_kernel(const float* __restrict__ Wt,
                                                   const float* __restrict__ bias,
                                                   const float* __restrict__ Gp,
                                                   const int*   __restrict__ meta,
                                                   float*       __restrict__ out) {
    constexpr int WAVES  = 8;
    constexpr int STRIDE = 4 * WAVES;           // 32 floats per wave step
    constexpr int NSTEP  = K_DIM / STRIDE;      // 1250 steps per wave
    constexpr int UF     = 5;                   // steps per chunk
    constexpr int NITER  = NSTEP / UF;          // 250 chunks (even)

    __shared__ float red[WAVES - 1][MT][8][32]; // partials of waves 1..7

    const int s     = blockIdx.y;
    const int count = meta[META_COUNT + s];
    if (((count + 15) >> 4) != MT) return;      // uniform; covers count==0
    const int start  = meta[META_START + s];
    const int n_base = blockIdx.x * 16;

    const int lane  = threadIdx.x & 31;
    const int wave  = threadIdx.x >> 5;
    const int nn    = lane & 15;                // N (and M for A loads)
    const int khalf = (lane >> 4) << 1;         // 0 | 2 : K sub-position

    // 32-bit element offsets (max W offset = 491 MB < 4 GiB) so the backend
    // uses saddr + 32-bit voffset + imm global addressing.
    uint32_t woff = (uint32_t)(((s * OUT_DIM) + n_base + nn) * K_DIM)
                  + (uint32_t)(khalf + wave * 4);
    uint32_t aoff[MT];
#pragma unroll
    for (int m = 0; m < MT; ++m) {
        const int gm = m * 16 + nn;
        const int rc = start + ((gm < count) ? gm : (count - 1));
        aoff[m] = (uint32_t)(rc * K_DIM) + (uint32_t)(khalf + wave * 4);
    }
    const float amul = ((MT - 1) * 16 + nn < count) ? 1.0f : 0.0f;

    v8f acc[MT];
#pragma unroll
    for (int m = 0; m < MT; ++m)
#pragma unroll
        for (int j = 0; j < 8; ++j) acc[m][j] = 0.0f;

    // --- two-buffer software pipeline over K -------------------------------
    v2f bt0[UF], bt1[UF];
    v2f at0[UF][MT], at1[UF][MT];

    auto load_chunk = [&](v2f (&bt)[UF], v2f (&at)[UF][MT]) {
#pragma unroll
        for (int u = 0; u < UF; ++u) {
            // stream W non-temporally: 491 MB must not evict L2-resident Gp
            bt[u] = __builtin_nontemporal_load(
                (const v2f*)(Wt + woff + (uint32_t)(u * STRIDE)));
#pragma unroll
            for (int m = 0; m < MT; ++m)
                at[u][m] = *(const v2f*)(Gp + aoff[m] + (uint32_t)(u * STRIDE));
        }
        woff += UF * STRIDE;
#pragma unroll
        for (int m = 0; m < MT; ++m) aoff[m] += UF * STRIDE;
    };

    auto drain_chunk = [&](const v2f (&bt)[UF], const v2f (&at)[UF][MT]) {
#pragma unroll
        for (int u = 0; u < UF; ++u) {
#pragma unroll
            for (int m = 0; m < MT; ++m) {
                v2f a = at[u][m];
                if (m == MT - 1) {              // only last tile is partial
                    a[0] *= amul;
                    a[1] *= amul;
                }
                acc[m] = __builtin_amdgcn_wmma_f32_16x16x4_f32(
                    false, a, false, bt[u], (short)0, acc[m], false, false);
            }
        }
    };

    load_chunk(bt0, at0);                       // prologue
    for (int it = 0; it < NITER / 2 - 1; ++it) {
        load_chunk(bt1, at1);
        drain_chunk(bt0, at0);
        load_chunk(bt0, at0);
        drain_chunk(bt1, at1);
    }
    load_chunk(bt1, at1);                       // epilogue
    drain_chunk(bt0, at0);
    drain_chunk(bt1, at1);
    // 1 + 124*2 + 1 = 250 loads, 124*2 + 2 = 250 drains  (all of K covered)

    // cross-wave reduction through LDS
    if (wave > 0) {
#pragma unroll
        for (int m = 0; m < MT; ++m)
#pragma unroll
            for (int j = 0; j < 8; ++j)
                red[wave - 1][m][j][lane] = acc[m][j];
    }
    __syncthreads();

    if (wave == 0) {
        const float bv = bias[s * OUT_DIM + n_base + nn];
#pragma unroll
        for (int m = 0; m < MT; ++m) {
#pragma unroll
            for (int j = 0; j < 8; ++j) {
                float v = acc[m][j];
#pragma unroll
                for (int w = 0; w < WAVES - 1; ++w) v += red[w][m][j][lane];
                const int mrow = j + ((lane >> 4) << 3);  // C layout: M index
                const int gm   = m * 16 + mrow;
                if (gm < count) {
                    const int bsmp = meta[META_PERM + start + gm];
                    out[(size_t)bsmp * OUT_DIM + n_base + nn] = v + bv;
                }
            }
        }
    }
}

// ---------------------------------------------------------------------------
// entry point
// ---------------------------------------------------------------------------
extern "C" void kernel_launch(void* const* d_in, const int* in_sizes, int n_in,
                              void* d_out, int out_size, void* d_ws, size_t ws_size,
                              hipStream_t stream) {
    const int*   id   = (const int*)d_in[0];    // id_batch   [64]
    const float* fmri = (const float*)d_in[1];  // fmri_batch [64,64,64,48]
    const int*   idx  = (const int*)d_in[2];    // idx        [4,40000]
    const float* Wt   = (const float*)d_in[3];  // Wt         [4,768,40000]
    const float* bias = (const float*)d_in[4];  // b          [4,768]
    float*       out  = (float*)d_out;          // [64,1,768]

    int*   meta = (int*)d_ws;
    float* Gp   = (float*)((char*)d_ws + GP_OFFSET_BYTES);

    group_kernel<<<1, 32, 0, stream>>>(id, meta);
    gather_kernel<<<dim3((K_DIM + 255) / 256, B_DIM), 256, 0, stream>>>(fmri, idx, meta, Gp);

    const dim3 ggrid(OUT_DIM / 16, S_DIM);
    gemm_kernel<1><<<ggrid, 256, 0, stream>>>(Wt, bias, Gp, meta, out);
    gemm_kernel<2><<<ggrid, 256, 0, stream>>>(Wt, bias, Gp, meta, out);
    gemm_kernel<3><<<ggrid, 256, 0, stream>>>(Wt, bias, Gp, meta, out);
    gemm_kernel<4><<<ggrid, 256, 0, stream>>>(Wt, bias, Gp, meta, out);
}